// GroupingTransformer_12086037971334
// MI455X (gfx1250) — compile-verified
//
#include <hip/hip_runtime.h>
#include <math.h>

#define B_ 4
#define N_ 512
#define D_ 128
#define H_ 4
#define HD_ 32
#define REL_ 64
#define FF_ 512
#define DEPTH_ 4
#define TOK_ (B_ * N_)   // 2048

typedef float v2f __attribute__((ext_vector_type(2)));
typedef float v8f __attribute__((ext_vector_type(8)));

// ---------------- WMMA helpers (f32 16x16x4, CDNA5 layouts) ----------------

static __device__ __forceinline__ v8f wmma4(v2f a, v2f b, v8f c) {
  // 8 args: (neg_a, A, neg_b, B, c_mod, C, reuse_a, reuse_b)
  return __builtin_amdgcn_wmma_f32_16x16x4_f32(false, a, false, b, (short)0, c,
                                               false, false);
}

// A tile 16x4 from row-major src[row][k], stride in floats.
// Lane l<16: m=l, k=0/1 ; lane l>=16: m=l-16, k=2/3.
static __device__ __forceinline__ v2f load_a16x4(const float* src, int stride,
                                                 int lane) {
  int m = lane & 15;
  int kb = (lane >> 4) << 1;
  v2f a;
  a.x = src[m * stride + kb];
  a.y = src[m * stride + kb + 1];
  return a;
}

// B tile 4x16 from row-major src[k][n], stride in floats.
// Lane l<16: n=l, k=0/1 ; lane l>=16: n=l-16, k=2/3.
static __device__ __forceinline__ v2f load_b4x16(const float* src, int stride,
                                                 int lane) {
  int n = lane & 15;
  int kb = (lane >> 4) << 1;
  v2f b;
  b.x = src[kb * stride + n];
  b.y = src[(kb + 1) * stride + n];
  return b;
}

static __device__ __forceinline__ float geluf(float x) {
  return 0.5f * x * (1.0f + erff(x * 0.70710678118654752f));
}

// ---------------- K0: concat per-layer Wrel/brel into (64x16)/(16) ----------

__global__ void build_wrel_kernel(const float* w0, const float* w1,
                                  const float* w2, const float* w3,
                                  const float* r0, const float* r1,
                                  const float* r2, const float* r3,
                                  float* WrelAll, float* brelAll) {
  int t = threadIdx.x;  // 64 threads = rows of Wrel
  const float* ws[4] = {w0, w1, w2, w3};
  for (int L = 0; L < DEPTH_; ++L)
    for (int hh = 0; hh < H_; ++hh)
      WrelAll[t * 16 + L * 4 + hh] = ws[L][t * 4 + hh];
  if (t < 16) {
    const float* rs[4] = {r0, r1, r2, r3};
    brelAll[t] = rs[t >> 2][t & 3];
  }
}

// ---------------- K1: object MLP  x = MLP([pos,size]) ------------------------

__global__ void obj_mlp_kernel(const float* __restrict__ pos,
                               const float* __restrict__ size,
                               const float* __restrict__ W1,
                               const float* __restrict__ b1,
                               const float* __restrict__ W2,
                               const float* __restrict__ b2,
                               float* __restrict__ x) {
  int row = blockIdx.x;  // token
  int t = threadIdx.x;   // 128 threads
  __shared__ float inp[3];
  __shared__ float hid[128];
  if (t < 2) inp[t] = pos[row * 2 + t];
  if (t == 2) inp[2] = size[row];
  __syncthreads();
  float a = b1[t];
  for (int k = 0; k < 3; ++k) a += inp[k] * W1[k * 128 + t];
  hid[t] = geluf(a);
  __syncthreads();
  float o = b2[t];
  for (int k = 0; k < 128; ++k) o += hid[k] * W2[k * 128 + t];
  x[row * 128 + t] = o;
}

// ---------------- K2: fused geom -> relMLP -> rbAll (B,N,N,16) ---------------
// One wave handles 16 pairs (fixed b,i ; 16 consecutive j).

#define RELW 4  // waves per block

__global__ void rel_rb_kernel(const float* __restrict__ pos,
                              const float* __restrict__ size,
                              const float* __restrict__ W1,
                              const float* __restrict__ b1,
                              const float* __restrict__ W2,
                              const float* __restrict__ b2,
                              const float* __restrict__ WrelAll,
                              const float* __restrict__ brelAll,
                              float* __restrict__ rbAll) {
  __shared__ float hidLDS[RELW][16][128];  // 32 KB
  __shared__ float relLDS[RELW][16][64];   // 16 KB
  int lane = threadIdx.x & 31;
  int wave = threadIdx.x >> 5;
  int job = blockIdx.x * RELW + wave;  // exact grid: B*N*(N/16) jobs
  int jt = job & 31;
  int i = (job >> 5) & (N_ - 1);
  int b = job >> 14;  // job / (32*512)
  int j0 = jt * 16;

  // geometry features for this lane's pair (A-matrix layout for 16x4)
  int t = lane & 15;
  int j = j0 + t;
  float pix = pos[(b * N_ + i) * 2 + 0], piy = pos[(b * N_ + i) * 2 + 1];
  float pjx = pos[(b * N_ + j) * 2 + 0], pjy = pos[(b * N_ + j) * 2 + 1];
  float dx = pix - pjx, dy = piy - pjy;
  float dist = sqrtf(dx * dx + dy * dy + 1e-6f);
  float sd = size[b * N_ + i] - size[b * N_ + j];
  v2f aGeom;
  if (lane < 16) { aGeom.x = dist; aGeom.y = dx; }   // k = 0,1
  else           { aGeom.x = dy;   aGeom.y = sd; }   // k = 2,3

  int n = lane & 15;
  int mb = (lane >> 4) << 3;
  float* hid = &hidLDS[wave][0][0];
  float* rel = &relLDS[wave][0][0];

  // stage 1: hidden(16x128) = gelu(geom(16x4) @ W1(4x128) + b1)
  // A fragment (aGeom) lives in registers, reused across all 8 N-tiles.
#pragma unroll
  for (int tn = 0; tn < 8; ++tn) {
    v2f bm = load_b4x16(W1 + tn * 16, 128, lane);
    v8f c = {};
    c = wmma4(aGeom, bm, c);
    float bv = b1[tn * 16 + n];
#pragma unroll
    for (int r = 0; r < 8; ++r)
      hid[(mb + r) * 128 + tn * 16 + n] = geluf(c[r] + bv);
  }
  __syncthreads();

  // stage 2: rel(16x64) = hidden @ W2(128x64) + b2
  // Register-blocked: one A load feeds 4 WMMAs (4 accumulators).
  {
    v8f c[4] = {v8f{}, v8f{}, v8f{}, v8f{}};
    for (int k0 = 0; k0 < 128; k0 += 4) {
      v2f a = load_a16x4(hid + k0, 128, lane);
#pragma unroll
      for (int tn = 0; tn < 4; ++tn) {
        v2f bm = load_b4x16(W2 + k0 * 64 + tn * 16, 64, lane);
        c[tn] = wmma4(a, bm, c[tn]);
      }
    }
#pragma unroll
    for (int tn = 0; tn < 4; ++tn) {
      float bv = b2[tn * 16 + n];
#pragma unroll
      for (int r = 0; r < 8; ++r)
        rel[(mb + r) * 64 + tn * 16 + n] = c[tn][r] + bv;
    }
  }
  __syncthreads();

  // stage 3: rb(16x16) = rel @ WrelAll(64x16) + brelAll -> global
  v8f c = {};
  for (int k0 = 0; k0 < 64; k0 += 4) {
    v2f a = load_a16x4(rel + k0, 64, lane);
    v2f bm = load_b4x16(WrelAll + k0 * 16, 16, lane);
    c = wmma4(a, bm, c);
  }
  float bv = brelAll[n];
#pragma unroll
  for (int r = 0; r < 8; ++r) {
    int jj = j0 + mb + r;
    rbAll[((long)(b * N_ + i) * N_ + jj) * 16 + n] = c[r] + bv;
  }
}

// ---------------- K3: LayerNorm ---------------------------------------------

__global__ void ln_kernel(const float* __restrict__ x,
                          const float* __restrict__ g,
                          const float* __restrict__ be,
                          float* __restrict__ y) {
  int row = blockIdx.x;
  int t = threadIdx.x;  // 128
  __shared__ float s1[128], s2[128];
  float v = x[row * 128 + t];
  s1[t] = v;
  s2[t] = v * v;
  __syncthreads();
  for (int off = 64; off; off >>= 1) {
    if (t < off) {
      s1[t] += s1[t + off];
      s2[t] += s2[t + off];
    }
    __syncthreads();
  }
  float mu = s1[0] * (1.0f / 128.0f);
  float var = s2[0] * (1.0f / 128.0f) - mu * mu;
  float rstd = rsqrtf(var + 1e-5f);
  y[row * 128 + t] = (v - mu) * rstd * g[t] + be[t];
}

// ---------------- K4: generic WMMA GEMM  Y = act(X@W + bias) [+ addTo] ------
// X: T x K row-major, W: K x Nf row-major.
// One wave per 16x64 output block (4 accumulators): one A load -> 4 WMMAs.
// Nf must be a multiple of 64.

__global__ void gemm_kernel(const float* __restrict__ X,
                            const float* __restrict__ W,
                            const float* __restrict__ bias,
                            const float* __restrict__ addTo,
                            float* __restrict__ Y, int T, int K, int Nf,
                            int act) {
  int lane = threadIdx.x & 31;
  int wave = threadIdx.x >> 5;
  int blk = blockIdx.x * (blockDim.x >> 5) + wave;
  int ng = Nf >> 6;  // groups of 4 n-tiles
  int total = (T >> 4) * ng;
  if (blk >= total) return;  // wave-uniform
  int tm = blk / ng, tg = blk % ng;

  const float* Xp = X + (long)tm * 16 * K;
  const float* Wp = W + tg * 64;
  v8f c[4] = {v8f{}, v8f{}, v8f{}, v8f{}};
  for (int k0 = 0; k0 < K; k0 += 4) {
    v2f a = load_a16x4(Xp + k0, K, lane);
#pragma unroll
    for (int u = 0; u < 4; ++u) {
      v2f b = load_b4x16(Wp + (long)k0 * Nf + u * 16, Nf, lane);
      c[u] = wmma4(a, b, c[u]);
    }
  }
  int n = lane & 15;
  int mb = (lane >> 4) << 3;
  float* Yp = Y + (long)(tm * 16) * Nf + tg * 64;
  const float* Ap = addTo ? addTo + (long)(tm * 16) * Nf + tg * 64 : nullptr;
#pragma unroll
  for (int u = 0; u < 4; ++u) {
    float bv = bias ? bias[tg * 64 + u * 16 + n] : 0.0f;
#pragma unroll
    for (int r = 0; r < 8; ++r) {
      float v = c[u][r] + bv;
      if (act) v = geluf(v);
      if (Ap) v += Ap[(mb + r) * Nf + u * 16 + n];
      Yp[(mb + r) * Nf + u * 16 + n] = v;
    }
  }
}

// ---------------- K5: fused attention (flash-style, per 16 query rows) ------

#define ATW 2  // waves per block -> 64 KB LDS

__global__ void attn_kernel(const float* __restrict__ qkv,
                            const float* __restrict__ rbAll,
                            float* __restrict__ o, int layer) {
  __shared__ float SL[ATW][16][N_];
  int lane = threadIdx.x & 31;
  int wave = threadIdx.x >> 5;
  int job = blockIdx.x * ATW + wave;  // exact: B*H*(N/16) = 512 jobs
  int it = job & 31;
  int h = (job >> 5) & 3;
  int b = job >> 7;
  int i0 = it * 16;
  float* S = &SL[wave][0][0];

  int n = lane & 15;
  int mb = (lane >> 4) << 3;
  int kb = (lane >> 4) << 1;
  const float scale = 0.17677669529663687f;  // 1/sqrt(32)

  // ---- preload q fragment (16x32): invariant across all key tiles ----
  const float* qbase = qkv + (long)(b * N_ + i0) * 384 + h * HD_;
  v2f qa[8];
#pragma unroll
  for (int k = 0; k < 8; ++k) {
    qa[k].x = qbase[n * 384 + k * 4 + kb];
    qa[k].y = qbase[n * 384 + k * 4 + kb + 1];
  }

  // ---- scores S = q k^T * scale + rb ----
  for (int jt = 0; jt < 32; ++jt) {
    int j0 = jt * 16;
    const float* kbase = qkv + (long)(b * N_ + j0) * 384 + 128 + h * HD_;
    v8f c = {};
#pragma unroll
    for (int k = 0; k < 8; ++k) {
      v2f bm;  // B = K^T : B[kk][nn] = K[j0+nn][kk]
      bm.x = kbase[n * 384 + k * 4 + kb];
      bm.y = kbase[n * 384 + k * 4 + kb + 1];
      c = wmma4(qa[k], bm, c);
    }
#pragma unroll
    for (int r = 0; r < 8; ++r) {
      int i = i0 + mb + r;
      int jcol = j0 + n;
      float v = c[r] * scale +
                rbAll[((long)(b * N_ + i) * N_ + jcol) * 16 + layer * 4 + h];
      S[(mb + r) * N_ + j0 + n] = v;
    }
  }
  __syncthreads();

  // ---- softmax over each of the 16 rows (wave-cooperative) ----
  for (int r = 0; r < 16; ++r) {
    float* row = S + r * N_;
    float mx = -1e30f;
    for (int jj = lane; jj < N_; jj += 32) mx = fmaxf(mx, row[jj]);
    for (int off = 16; off; off >>= 1) mx = fmaxf(mx, __shfl_xor(mx, off, 32));
    float sum = 0.0f;
    for (int jj = lane; jj < N_; jj += 32) {
      float e = expf(row[jj] - mx);
      row[jj] = e;
      sum += e;
    }
    for (int off = 16; off; off >>= 1) sum += __shfl_xor(sum, off, 32);
    float inv = 1.0f / sum;
    for (int jj = lane; jj < N_; jj += 32) row[jj] *= inv;
  }
  __syncthreads();

  // ---- O = P @ V  (16 x 32): share A fragment across both d-tiles ----
  {
    const float* vb0 = qkv + (long)b * N_ * 384 + 256 + h * HD_;
    const float* vb1 = vb0 + 16;
    v8f c0 = {}, c1 = {};
    for (int k0 = 0; k0 < N_; k0 += 4) {
      v2f a = load_a16x4(S + k0, N_, lane);
      v2f bm0, bm1;
      bm0.x = vb0[(k0 + kb) * 384 + n];
      bm0.y = vb0[(k0 + kb + 1) * 384 + n];
      bm1.x = vb1[(k0 + kb) * 384 + n];
      bm1.y = vb1[(k0 + kb + 1) * 384 + n];
      c0 = wmma4(a, bm0, c0);
      c1 = wmma4(a, bm1, c1);
    }
#pragma unroll
    for (int r = 0; r < 8; ++r) {
      o[(long)(b * N_ + i0 + mb + r) * 128 + h * HD_ + n] = c0[r];
      o[(long)(b * N_ + i0 + mb + r) * 128 + h * HD_ + 16 + n] = c1[r];
    }
  }
}

// ---------------- K6: fused pairwise affinity -------------------------------

__global__ void aff_kernel(const float* __restrict__ U,
                           const float* __restrict__ V,
                           const float* __restrict__ ba1,
                           const float* __restrict__ Wa2,
                           const float* __restrict__ ba2p,
                           float* __restrict__ raw) {
  int bi = blockIdx.x;  // b*N + i
  int b = bi >> 9;
  int t = threadIdx.x;  // 256
  __shared__ float u[128], bb[128], w2[128];
  if (t < 128) {
    u[t] = U[(long)bi * 128 + t];
    bb[t] = ba1[t];
    w2[t] = Wa2[t];
  }
  __syncthreads();
  float ba2 = ba2p[0];
  for (int j = t; j < N_; j += 256) {
    const float* vr = V + ((long)b * N_ + j) * 128;
    float acc = ba2;
    for (int d = 0; d < 128; ++d) acc += geluf(u[d] + vr[d] + bb[d]) * w2[d];
    raw[(long)bi * N_ + j] = acc;
  }
}

// ---------------- K7: symmetrize --------------------------------------------

__global__ void symm_kernel(const float* __restrict__ raw,
                            float* __restrict__ out) {
  long idx = (long)blockIdx.x * blockDim.x + threadIdx.x;
  int j = idx & (N_ - 1);
  int i = (idx >> 9) & (N_ - 1);
  int b = idx >> 18;
  out[idx] = 0.5f * (raw[((long)(b * N_ + i)) * N_ + j] +
                     raw[((long)(b * N_ + j)) * N_ + i]);
}

// ---------------- host launcher ---------------------------------------------

extern "C" void kernel_launch(void* const* d_in, const int* in_sizes, int n_in,
                              void* d_out, int out_size, void* d_ws,
                              size_t ws_size, hipStream_t stream) {
  (void)in_sizes; (void)n_in; (void)out_size; (void)ws_size;
  const float* pos = (const float*)d_in[0];
  const float* size = (const float*)d_in[1];
  const float* objW1 = (const float*)d_in[2];
  const float* objB1 = (const float*)d_in[3];
  const float* objW2 = (const float*)d_in[4];
  const float* objB2 = (const float*)d_in[5];
  const float* relW1 = (const float*)d_in[6];
  const float* relB1 = (const float*)d_in[7];
  const float* relW2 = (const float*)d_in[8];
  const float* relB2 = (const float*)d_in[9];
  struct LayerP {
    const float *g1, *b1, *Wqkv, *bqkv, *Wout, *bout, *Wrel, *brel, *g2, *b2,
        *Wm1, *bm1, *Wm2, *bm2;
  } lp[DEPTH_];
  int idx = 10;
  for (int L = 0; L < DEPTH_; ++L) {
    lp[L].g1 = (const float*)d_in[idx++];
    lp[L].b1 = (const float*)d_in[idx++];
    lp[L].Wqkv = (const float*)d_in[idx++];
    lp[L].bqkv = (const float*)d_in[idx++];
    lp[L].Wout = (const float*)d_in[idx++];
    lp[L].bout = (const float*)d_in[idx++];
    lp[L].Wrel = (const float*)d_in[idx++];
    lp[L].brel = (const float*)d_in[idx++];
    lp[L].g2 = (const float*)d_in[idx++];
    lp[L].b2 = (const float*)d_in[idx++];
    lp[L].Wm1 = (const float*)d_in[idx++];
    lp[L].bm1 = (const float*)d_in[idx++];
    lp[L].Wm2 = (const float*)d_in[idx++];
    lp[L].bm2 = (const float*)d_in[idx++];
  }
  const float* Wa1 = (const float*)d_in[idx++];
  const float* ba1 = (const float*)d_in[idx++];
  const float* Wa2 = (const float*)d_in[idx++];
  const float* ba2 = (const float*)d_in[idx++];

  // workspace layout (floats)
  float* ws = (float*)d_ws;
  float* rbAll = ws;            size_t off = (size_t)B_ * N_ * N_ * 16;  // 64MB
  float* x = ws + off;          off += TOK_ * D_;
  float* h = ws + off;          off += TOK_ * D_;
  float* qkv = ws + off;        off += TOK_ * 3 * D_;
  float* ovec = ws + off;       off += TOK_ * D_;
  float* ff = ws + off;         off += TOK_ * FF_;
  float* U = ws + off;          off += TOK_ * D_;
  float* V = ws + off;          off += TOK_ * D_;
  float* WrelAll = ws + off;    off += REL_ * 16;
  float* brelAll = ws + off;    off += 16;
  float* raw = ws + off;        off += (size_t)B_ * N_ * N_;

  build_wrel_kernel<<<1, 64, 0, stream>>>(
      lp[0].Wrel, lp[1].Wrel, lp[2].Wrel, lp[3].Wrel, lp[0].brel, lp[1].brel,
      lp[2].brel, lp[3].brel, WrelAll, brelAll);

  obj_mlp_kernel<<<TOK_, 128, 0, stream>>>(pos, size, objW1, objB1, objW2,
                                           objB2, x);

  // B*N*(N/16) = 65536 wave-jobs / RELW waves per block
  rel_rb_kernel<<<(B_ * N_ * (N_ / 16)) / RELW, RELW * 32, 0, stream>>>(
      pos, size, relW1, relB1, relW2, relB2, WrelAll, brelAll, rbAll);

  for (int L = 0; L < DEPTH_; ++L) {
    ln_kernel<<<TOK_, 128, 0, stream>>>(x, lp[L].g1, lp[L].b1, h);
    gemm_kernel<<<(TOK_ / 16) * (384 / 64) / 4, 128, 0, stream>>>(
        h, lp[L].Wqkv, lp[L].bqkv, nullptr, qkv, TOK_, D_, 3 * D_, 0);
    attn_kernel<<<(B_ * H_ * (N_ / 16)) / ATW, ATW * 32, 0, stream>>>(
        qkv, rbAll, ovec, L);
    gemm_kernel<<<(TOK_ / 16) * (128 / 64) / 4, 128, 0, stream>>>(
        ovec, lp[L].Wout, lp[L].bout, x, x, TOK_, D_, D_, 0);
    ln_kernel<<<TOK_, 128, 0, stream>>>(x, lp[L].g2, lp[L].b2, h);
    gemm_kernel<<<(TOK_ / 16) * (FF_ / 64) / 4, 128, 0, stream>>>(
        h, lp[L].Wm1, lp[L].bm1, nullptr, ff, TOK_, D_, FF_, 1);
    gemm_kernel<<<(TOK_ / 16) * (128 / 64) / 4, 128, 0, stream>>>(
        ff, lp[L].Wm2, lp[L].bm2, x, x, TOK_, FF_, D_, 0);
  }

  // U = x @ Wa1[:128], V = x @ Wa1[128:]
  gemm_kernel<<<(TOK_ / 16) * (128 / 64) / 4, 128, 0, stream>>>(
      x, Wa1, nullptr, nullptr, U, TOK_, D_, D_, 0);
  gemm_kernel<<<(TOK_ / 16) * (128 / 64) / 4, 128, 0, stream>>>(
      x, Wa1 + 128 * 128, nullptr, nullptr, V, TOK_, D_, D_, 0);

  aff_kernel<<<TOK_, 256, 0, stream>>>(U, V, ba1, Wa2, ba2, raw);
  symm_kernel<<<(B_ * N_ * N_) / 256, 256, 0, stream>>>(raw, (float*)d_out);
}